// VariationalGCNEncoder_15539191677587
// MI455X (gfx1250) — compile-verified
//
#include <hip/hip_runtime.h>

typedef float v2f __attribute__((ext_vector_type(2)));
typedef float v8f __attribute__((ext_vector_type(8)));

#define IN_C 128
#define HID  64
#define OUTC 32

// ---------------- degree / normalization ----------------

__global__ void k_init_deg(float* __restrict__ deg, int n) {
    int i = blockIdx.x * blockDim.x + threadIdx.x;
    if (i < n) deg[i] = 1.0f;               // self-loop
}

__global__ void k_count_deg(const int* __restrict__ dst, float* __restrict__ deg, int e) {
    int i = blockIdx.x * blockDim.x + threadIdx.x;
    if (i < e) atomicAdd(&deg[dst[i]], 1.0f);
}

__global__ void k_rsqrt_inplace(float* __restrict__ deg, int n) {
    int i = blockIdx.x * blockDim.x + threadIdx.x;
    if (i < n) deg[i] = rsqrtf(deg[i]);     // deg buffer becomes dinv
}

// ---------------- GEMM 1: lin1 = x @ W1  (N x 128 x 64), fp32 WMMA ----------------
// One wave per 16x64 output slab: A fragment loaded once per K-step, reused by
// 4 independent WMMAs (4 accumulators -> no RAW hazard, they pipeline).
// A 16x4 f32 layout: lane l holds A[l&15, 2*(l>>4)+v]; B symmetric; C/D: v8f,
// VGPR i = rows i and i+8, N = lane&15.

__global__ __launch_bounds__(256) void k_gemm1(const float* __restrict__ x,
                                               const float* __restrict__ W1,
                                               float* __restrict__ lin1, int n) {
    int wave = blockIdx.x * (blockDim.x >> 5) + (threadIdx.x >> 5);
    if (wave * 16 >= n) return;             // wave-uniform: EXEC stays all-1s
    int lane = threadIdx.x & 31;
    int r  = lane & 15;
    int kb = (lane >> 4) << 1;              // 0 or 2
    int arow = wave * 16 + r;
    v8f acc0 = {}, acc1 = {}, acc2 = {}, acc3 = {};
    #pragma unroll 4
    for (int k0 = 0; k0 < IN_C; k0 += 4) {
        v2f a;
        a.x = x[arow * IN_C + k0 + kb];
        a.y = x[arow * IN_C + k0 + kb + 1];
        const float* wp = W1 + (k0 + kb) * HID + r;
        v2f b0, b1, b2, b3;
        b0.x = wp[0];  b0.y = wp[HID];
        b1.x = wp[16]; b1.y = wp[HID + 16];
        b2.x = wp[32]; b2.y = wp[HID + 32];
        b3.x = wp[48]; b3.y = wp[HID + 48];
        acc0 = __builtin_amdgcn_wmma_f32_16x16x4_f32(false, a, false, b0, (short)0, acc0, false, false);
        acc1 = __builtin_amdgcn_wmma_f32_16x16x4_f32(false, a, false, b1, (short)0, acc1, false, false);
        acc2 = __builtin_amdgcn_wmma_f32_16x16x4_f32(false, a, false, b2, (short)0, acc2, false, false);
        acc3 = __builtin_amdgcn_wmma_f32_16x16x4_f32(false, a, false, b3, (short)0, acc3, false, false);
    }
    int mb = (lane >> 4) * 8;
    float* o = lin1 + (wave * 16 + mb) * HID + r;
    #pragma unroll
    for (int i = 0; i < 8; ++i) {
        o[i * HID +  0] = acc0[i];
        o[i * HID + 16] = acc1[i];
        o[i * HID + 32] = acc2[i];
        o[i * HID + 48] = acc3[i];
    }
}

// ---------------- aggregation init: agg1 = lin1 * dinv^2 + b1 ----------------

__global__ void k_init_agg1(const float* __restrict__ lin1, const float* __restrict__ dinv,
                            const float* __restrict__ b1, float* __restrict__ agg1, int n) {
    int idx = blockIdx.x * blockDim.x + threadIdx.x;
    if (idx >= n * HID) return;
    int node = idx >> 6;
    int c = idx & 63;
    float di = dinv[node];
    agg1[idx] = lin1[idx] * (di * di) + b1[c];
}

// ---------------- edge scatter: agg1[dst] += lin1[src] * dinv[src]*dinv[dst] ----------------
// 16 threads per edge, 4 channels each (float4 gather, 4 scalar f32 atomics).

__global__ void k_scatter1(const int* __restrict__ src, const int* __restrict__ dst,
                           const float* __restrict__ dinv, const float* __restrict__ lin1,
                           float* __restrict__ agg1, int e) {
    int idx = blockIdx.x * blockDim.x + threadIdx.x;
    int ei = idx >> 4;
    if (ei >= e) return;
    int c = (idx & 15) << 2;
    int s = src[ei], d = dst[ei];
    float norm = dinv[s] * dinv[d];
    float4 v = *reinterpret_cast<const float4*>(lin1 + s * HID + c);
    float* p = agg1 + d * HID + c;
    atomicAdd(p + 0, v.x * norm);
    atomicAdd(p + 1, v.y * norm);
    atomicAdd(p + 2, v.z * norm);
    atomicAdd(p + 3, v.w * norm);
}

// ---------------- GEMM 2: lin2 = relu(agg1) @ [W_mu | W_logstd]  (N x 64 x 64) ----------------
// ReLU fused into the A-operand load; one wave per 16x64 slab (cols 0-31 = mu,
// 32-63 = logstd).

__global__ __launch_bounds__(256) void k_gemm2(const float* __restrict__ agg1,
                                               const float* __restrict__ Wmu,
                                               const float* __restrict__ Wls,
                                               float* __restrict__ lin2, int n) {
    int wave = blockIdx.x * (blockDim.x >> 5) + (threadIdx.x >> 5);
    if (wave * 16 >= n) return;
    int lane = threadIdx.x & 31;
    int r  = lane & 15;
    int kb = (lane >> 4) << 1;
    int arow = wave * 16 + r;
    v8f acc0 = {}, acc1 = {}, acc2 = {}, acc3 = {};
    #pragma unroll 4
    for (int k0 = 0; k0 < HID; k0 += 4) {
        v2f a;
        a.x = fmaxf(agg1[arow * HID + k0 + kb],     0.0f);
        a.y = fmaxf(agg1[arow * HID + k0 + kb + 1], 0.0f);
        const float* wm = Wmu + (k0 + kb) * OUTC + r;
        const float* wl = Wls + (k0 + kb) * OUTC + r;
        v2f b0, b1, b2, b3;
        b0.x = wm[0];  b0.y = wm[OUTC];
        b1.x = wm[16]; b1.y = wm[OUTC + 16];
        b2.x = wl[0];  b2.y = wl[OUTC];
        b3.x = wl[16]; b3.y = wl[OUTC + 16];
        acc0 = __builtin_amdgcn_wmma_f32_16x16x4_f32(false, a, false, b0, (short)0, acc0, false, false);
        acc1 = __builtin_amdgcn_wmma_f32_16x16x4_f32(false, a, false, b1, (short)0, acc1, false, false);
        acc2 = __builtin_amdgcn_wmma_f32_16x16x4_f32(false, a, false, b2, (short)0, acc2, false, false);
        acc3 = __builtin_amdgcn_wmma_f32_16x16x4_f32(false, a, false, b3, (short)0, acc3, false, false);
    }
    int mb = (lane >> 4) * 8;
    float* o = lin2 + (wave * 16 + mb) * HID + r;
    #pragma unroll
    for (int i = 0; i < 8; ++i) {
        o[i * HID +  0] = acc0[i];   // mu cols 0-15
        o[i * HID + 16] = acc1[i];   // mu cols 16-31
        o[i * HID + 32] = acc2[i];   // logstd cols 0-15
        o[i * HID + 48] = acc3[i];   // logstd cols 16-31
    }
}

// ---------------- output init: d_out = lin2 * dinv^2 + bias, split mu/logstd ----------------

__global__ void k_init_out(const float* __restrict__ lin2, const float* __restrict__ dinv,
                           const float* __restrict__ bmu, const float* __restrict__ bls,
                           float* __restrict__ out, int n) {
    int idx = blockIdx.x * blockDim.x + threadIdx.x;
    if (idx >= n * HID) return;
    int node = idx >> 6;
    int c = idx & 63;
    float di = dinv[node];
    float v = lin2[idx] * (di * di);
    if (c < OUTC) out[node * OUTC + c] = v + bmu[c];
    else          out[n * OUTC + node * OUTC + (c - OUTC)] = v + bls[c - OUTC];
}

// ---------------- edge scatter 2: into mu region and logstd region of d_out ----------------

__global__ void k_scatter2(const int* __restrict__ src, const int* __restrict__ dst,
                           const float* __restrict__ dinv, const float* __restrict__ lin2,
                           float* __restrict__ out, int n, int e) {
    int idx = blockIdx.x * blockDim.x + threadIdx.x;
    int ei = idx >> 4;
    if (ei >= e) return;
    int c = (idx & 15) << 2;                   // 4-aligned, never straddles c==32
    int s = src[ei], d = dst[ei];
    float norm = dinv[s] * dinv[d];
    float4 v = *reinterpret_cast<const float4*>(lin2 + s * HID + c);
    float* p = (c < OUTC) ? (out + d * OUTC + c)
                          : (out + n * OUTC + d * OUTC + (c - OUTC));
    atomicAdd(p + 0, v.x * norm);
    atomicAdd(p + 1, v.y * norm);
    atomicAdd(p + 2, v.z * norm);
    atomicAdd(p + 3, v.w * norm);
}

// ---------------- launch ----------------

extern "C" void kernel_launch(void* const* d_in, const int* in_sizes, int n_in,
                              void* d_out, int out_size, void* d_ws, size_t ws_size,
                              hipStream_t stream) {
    const float* x    = (const float*)d_in[0];
    const int*   eidx = (const int*)d_in[1];
    const float* W1   = (const float*)d_in[2];
    const float* b1   = (const float*)d_in[3];
    const float* Wmu  = (const float*)d_in[4];
    const float* bmu  = (const float*)d_in[5];
    const float* Wls  = (const float*)d_in[6];
    const float* bls  = (const float*)d_in[7];
    float* out = (float*)d_out;

    const int n = in_sizes[0] / IN_C;     // 100000
    const int e = in_sizes[1] / 2;        // 1600000
    const int* src = eidx;
    const int* dst = eidx + e;

    // workspace layout (fp32), all 16B aligned since n*4 % 16 == 0
    float* dinv = (float*)d_ws;           // n
    float* lin1 = dinv + n;               // n*64
    float* agg1 = lin1 + (size_t)n * HID; // n*64
    float* lin2 = agg1 + (size_t)n * HID; // n*64

    const int B = 256;
    // normalization
    k_init_deg<<<(n + B - 1) / B, B, 0, stream>>>(dinv, n);
    k_count_deg<<<(e + B - 1) / B, B, 0, stream>>>(dst, dinv, e);
    k_rsqrt_inplace<<<(n + B - 1) / B, B, 0, stream>>>(dinv, n);

    // layer 1
    const int waves = (n + 15) / 16;                // one wave per 16-row slab
    k_gemm1<<<(waves + 7) / 8, B, 0, stream>>>(x, W1, lin1, n);
    k_init_agg1<<<((size_t)n * HID + B - 1) / B, B, 0, stream>>>(lin1, dinv, b1, agg1, n);
    k_scatter1<<<(((size_t)e * 16) + B - 1) / B, B, 0, stream>>>(src, dst, dinv, lin1, agg1, e);

    // layer 2 (mu & logstd fused, relu fused into GEMM A-load)
    k_gemm2<<<(waves + 7) / 8, B, 0, stream>>>(agg1, Wmu, Wls, lin2, n);
    k_init_out<<<((size_t)n * HID + B - 1) / B, B, 0, stream>>>(lin2, dinv, bmu, bls, out, n);
    k_scatter2<<<(((size_t)e * 16) + B - 1) / B, B, 0, stream>>>(src, dst, dinv, lin2, out, n, e);
}